// HyperbolicGraphConvolution_86990267613554
// MI455X (gfx1250) — compile-verified
//
#include <hip/hip_runtime.h>
#include <math.h>

typedef float v2f __attribute__((ext_vector_type(2)));
typedef float v8f __attribute__((ext_vector_type(8)));

#define MIN_NORM 1e-15f
#define MAXNORM  0.996f      /* (1 - PROJ_EPS)/SQC */
#define ART_EPS  1e-7f

__device__ __forceinline__ float artanh_clip(float z) {
    z = fminf(fmaxf(z, -1.0f + ART_EPS), 1.0f - ART_EPS);
    return 0.5f * logf((1.0f + z) / (1.0f - z));
}
__device__ __forceinline__ float sigmoidf(float z) {
    return 1.0f / (1.0f + expf(-z));
}

// ---- CDNA5 async-copy helpers (inline asm; CDNA5 ISA) ----------
__device__ __forceinline__ void wait_async0() {
    asm volatile("s_wait_asynccnt 0x0" ::: "memory");
}
__device__ __forceinline__ void wait_ds0() {
    asm volatile("s_wait_dscnt 0x0" ::: "memory");
}
// Stage one contiguous 8KB tile (16 rows x 128 f32) global -> LDS.
// 32 lanes x 16 issues x 16B, ASYNCcnt-tracked, no VGPR data round-trip.
__device__ __forceinline__ void issue_tile_async(const float* gsrc, float* ldst,
                                                 int lane) {
#pragma unroll
    for (int it = 0; it < 16; ++it) {
        const int fo = (it * 32 + lane) * 4;                 // float offset (16B chunk)
        const unsigned int       laddr = (unsigned int)(uintptr_t)(ldst + fo);
        const unsigned long long gaddr = (unsigned long long)(uintptr_t)(gsrc + fo);
        asm volatile("global_load_async_to_lds_b128 %0, %1, off"
                     :: "v"(laddr), "v"(gaddr)
                     : "memory");
    }
}

// 128-thread block sum, broadcast result to all threads.
__device__ __forceinline__ float block_sum(float v, float* s) {
    int t = threadIdx.x;
    s[t] = v; __syncthreads();
#pragma unroll
    for (int off = 64; off > 0; off >>= 1) {
        if (t < off) s[t] += s[t + off];
        __syncthreads();
    }
    float r = s[0]; __syncthreads();
    return r;
}

// ---------------- K1: mv = x (1024x128) @ W^T (128x128) via f32 WMMA ----------
__global__ __launch_bounds__(32)
void k_linear_wmma(const float* __restrict__ x, const float* __restrict__ w,
                   float* __restrict__ mv) {
    const int lane = threadIdx.x;
    const int lrow = lane & 15, hi = lane >> 4;
    const int m0 = blockIdx.x * 16, n0 = blockIdx.y * 16;
    v8f c = {0.f,0.f,0.f,0.f,0.f,0.f,0.f,0.f};
#pragma unroll 4
    for (int kk = 0; kk < 32; ++kk) {
        const int k = kk * 4 + 2 * hi;
        v2f a, b;
        a.x = x[(m0 + lrow) * 128 + k];     a.y = x[(m0 + lrow) * 128 + k + 1];
        b.x = w[(n0 + lrow) * 128 + k];     b.y = w[(n0 + lrow) * 128 + k + 1]; // B[k][n]=W[n][k]
        c = __builtin_amdgcn_wmma_f32_16x16x4_f32(false, a, false, b, (short)0, c,
                                                  false, false);
    }
    union { v8f v; float f[8]; } u; u.v = c;
#pragma unroll
    for (int v = 0; v < 8; ++v)
        mv[(m0 + v + 8 * hi) * 128 + n0 + lrow] = u.f[v];
}

// ---------------- K2: rowwise hyperbolic prep -> h, x2, tl, tr ----------------
__global__ __launch_bounds__(128)
void k_rowprep(const float* __restrict__ x, const float* __restrict__ mv,
               const float* __restrict__ bias, const float* __restrict__ att_w,
               float* __restrict__ h, float* __restrict__ x2a,
               float* __restrict__ tla, float* __restrict__ tra) {
    __shared__ float sred[128];
    const int r = blockIdx.x, t = threadIdx.x;
    float xv  = x[r * 128 + t];
    float mvv = mv[r * 128 + t];

    // mobius_matvec tail
    float xn2 = block_sum(xv * xv, sred);
    float mn2 = block_sum(mvv * mvv, sred);
    float asm_ = block_sum(fabsf(mvv), sred);
    float xn  = fmaxf(sqrtf(xn2), MIN_NORM);
    float mxn = fmaxf(sqrtf(mn2), MIN_NORM);
    float res = tanhf(mxn / xn * artanh_clip(xn)) * mvv / mxn;
    if (asm_ == 0.0f) res = 0.0f;
    float rn = fmaxf(sqrtf(block_sum(res * res, sred)), MIN_NORM);
    if (rn > MAXNORM) res = res / rn * MAXNORM;

    // hyp_bias = proj(expmap0(bias))
    float bb = bias[t];
    float bn = fmaxf(sqrtf(block_sum(bb * bb, sred)), MIN_NORM);
    float hb = tanhf(bn) * bb / bn;
    float hbn = fmaxf(sqrtf(block_sum(hb * hb, sred)), MIN_NORM);
    if (hbn > MAXNORM) hb = hb / hbn * MAXNORM;

    // h = proj(mobius_add(res, hb))
    float ax2 = block_sum(res * res, sred);
    float by2 = block_sum(hb * hb, sred);
    float ab  = block_sum(res * hb, sred);
    float den = fmaxf(1.0f + 2.0f * ab + ax2 * by2, MIN_NORM);
    float hv  = ((1.0f + 2.0f * ab + by2) * res + (1.0f - ax2) * hb) / den;
    float hn  = fmaxf(sqrtf(block_sum(hv * hv, sred)), MIN_NORM);
    if (hn > MAXNORM) hv = hv / hn * MAXNORM;

    float x2r = block_sum(hv * hv, sred);
    float hnc = fmaxf(sqrtf(x2r), MIN_NORM);
    float xt  = artanh_clip(hnc) * hv / hnc;     // logmap0(h)
    float tl  = block_sum(xt * att_w[t], sred);
    float tr  = block_sum(xt * att_w[128 + t], sred);

    h[r * 128 + t] = hv;
    if (t == 0) { x2a[r] = x2r; tla[r] = tl; tra[r] = tr; }
}

// ------- K3: fused Gram-WMMA -> attention/logmap scalars -> aggregation-WMMA --
// Double-buffered async global->LDS staging of the h j-tile overlaps the WMMAs.
__global__ __launch_bounds__(32)
void k_att_agg(const float* __restrict__ h, const float* __restrict__ adj,
               const float* __restrict__ x2a, const float* __restrict__ tla,
               const float* __restrict__ tra, const float* __restrict__ att_b,
               float* __restrict__ supp) {
    __shared__ float Ht[2][2048];                // double-buffered 16x128 h j-tile
    __shared__ float Ut[256];                    // 16x16 u_ij tile (transpose stage)
    const int lane = threadIdx.x;
    const int lrow = lane & 15, hi = lane >> 4;
    const int i0 = blockIdx.x * 16;
    const float attb = att_b[0];

    // Preload A fragments of h rows i0..i0+15 (K = full 128)
    v2f aH[32];
#pragma unroll
    for (int kk = 0; kk < 32; ++kk) {
        const int k = kk * 4 + 2 * hi;
        aH[kk].x = h[(i0 + lrow) * 128 + k];
        aH[kk].y = h[(i0 + lrow) * 128 + k + 1];
    }
    float x2i[8], tli[8], bti[8], sreg[8];
#pragma unroll
    for (int v = 0; v < 8; ++v) {
        const int row = i0 + v + 8 * hi;
        x2i[v] = x2a[row]; tli[v] = tla[row];
        bti[v] = 1.0f - x2i[v]; sreg[v] = 0.0f;
    }
    v8f acc[8];
#pragma unroll
    for (int nt = 0; nt < 8; ++nt) acc[nt] = (v8f){0.f,0.f,0.f,0.f,0.f,0.f,0.f,0.f};

    // Prologue: stage tile 0
    issue_tile_async(h, Ht[0], lane);
    int buf = 0;

    for (int jt = 0; jt < 64; ++jt) {
        const int j0 = jt * 16;
        wait_async0();                            // current buffer resident
        if (jt + 1 < 64) {
            wait_ds0();                           // drain reads of the other buffer
            issue_tile_async(h + (jt + 1) * 2048, Ht[buf ^ 1], lane);
        }
        const float* ht = Ht[buf];

        // ---- Gram tile G = h_i . h_j (16x16, K=128), B-frags from LDS ----
        v8f g = {0.f,0.f,0.f,0.f,0.f,0.f,0.f,0.f};
        for (int kk = 0; kk < 32; ++kk) {
            const int k = kk * 4 + 2 * hi;
            v2f b;
            b.x = ht[lrow * 128 + k];
            b.y = ht[lrow * 128 + k + 1];
            g = __builtin_amdgcn_wmma_f32_16x16x4_f32(false, aH[kk], false, b,
                                                      (short)0, g, false, false);
        }
        union { v8f v; float f[8]; } gu; gu.v = g;
        const int jn  = j0 + lrow;               // this lane's j column
        const float x2j = x2a[jn];
        const float trj = tra[jn];
#pragma unroll
        for (int v = 0; v < 8; ++v) {
            const int mrow = v + 8 * hi;
            const int i = i0 + mrow;
            const float G   = gu.f[v];
            const float a   = __builtin_nontemporal_load(&adj[i * 1024 + jn]);
            const float att = sigmoidf(tli[v] + trj + attb);
            const float alpha = 1.0f - 2.0f * G + x2j;         // coeff of -h_i
            const float beta  = bti[v];                        // coeff of  h_j
            const float den = fmaxf(1.0f - 2.0f * G + x2i[v] * x2j, MIN_NORM);
            const float s2  = (alpha * alpha * x2i[v] - 2.0f * alpha * beta * G
                               + beta * beta * x2j) / (den * den);
            const float n   = fmaxf(sqrtf(fmaxf(s2, 0.0f)), MIN_NORM);
            const float f   = bti[v] * artanh_clip(n) / n;     // 2/(c·lam)·artanh(n)/n
            const float uij = a * att * f / den;
            Ut[mrow * 16 + lrow] = uij;                        // ds_store (transpose stage)
            // s_i -= u_ij * alpha : shuffle-reduce over the 16 lanes of this hi-group
            float part = -uij * alpha;
#pragma unroll
            for (int m = 1; m < 16; m <<= 1) part += __shfl_xor(part, m, 32);
            sreg[v] += part;                                   // same value in all 16 lanes
        }
        __syncthreads();
        // ---- aggregation: acc(16x128) += U(16x16) @ h_jtile(16x128), all from LDS ----
        v2f ua[4];
#pragma unroll
        for (int kk = 0; kk < 4; ++kk) {                       // ds_load A fragments
            const int k = kk * 4 + 2 * hi;
            ua[kk].x = Ut[lrow * 16 + k];
            ua[kk].y = Ut[lrow * 16 + k + 1];
        }
#pragma unroll
        for (int nt = 0; nt < 8; ++nt) {
            v8f c = acc[nt];
#pragma unroll
            for (int kk = 0; kk < 4; ++kk) {
                const int kr = kk * 4 + 2 * hi;                // local k row in tile
                v2f b;
                b.x = ht[kr * 128 + nt * 16 + lrow];
                b.y = ht[(kr + 1) * 128 + nt * 16 + lrow];
                c = __builtin_amdgcn_wmma_f32_16x16x4_f32(false, ua[kk], false, b,
                                                          (short)0, c, false, false);
            }
            acc[nt] = c;
        }
        __syncthreads();
        buf ^= 1;
    }
    // supp = s_i * h_i + beta_i * acc
#pragma unroll
    for (int nt = 0; nt < 8; ++nt) {
        union { v8f v; float f[8]; } cu; cu.v = acc[nt];
#pragma unroll
        for (int v = 0; v < 8; ++v) {
            const int row = i0 + v + 8 * hi;
            const int col = nt * 16 + lrow;
            supp[row * 128 + col] = sreg[v] * h[row * 128 + col] + bti[v] * cu.f[v];
        }
    }
}

// ---------------- K4: out = proj(expmap0(proj(expmap(h, supp)))) --------------
__global__ __launch_bounds__(128)
void k_finish(const float* __restrict__ h, const float* __restrict__ supp,
              const float* __restrict__ x2a, float* __restrict__ out) {
    __shared__ float sred[128];
    const int r = blockIdx.x, t = threadIdx.x;
    float hv = h[r * 128 + t];
    float uv = supp[r * 128 + t];
    float x2r = x2a[r];

    float un  = fmaxf(sqrtf(block_sum(uv * uv, sred)), MIN_NORM);
    float lam = 2.0f / fmaxf(1.0f - x2r, MIN_NORM);
    float sec = tanhf(lam * un * 0.5f) * uv / un;

    float y2 = block_sum(sec * sec, sred);
    float xy = block_sum(hv * sec, sred);
    float den = fmaxf(1.0f + 2.0f * xy + x2r * y2, MIN_NORM);
    float ha = ((1.0f + 2.0f * xy + y2) * hv + (1.0f - x2r) * sec) / den;

    float han = fmaxf(sqrtf(block_sum(ha * ha, sred)), MIN_NORM);
    if (han > MAXNORM) ha = ha / han * MAXNORM;

    float na = fmaxf(sqrtf(block_sum(ha * ha, sred)), MIN_NORM);
    float o  = tanhf(na) * ha / na;                 // expmap0
    float on = fmaxf(tanhf(na), MIN_NORM);          // exact ||o||
    if (on > MAXNORM) o = o / on * MAXNORM;
    out[r * 128 + t] = o;
}

extern "C" void kernel_launch(void* const* d_in, const int* in_sizes, int n_in,
                              void* d_out, int out_size, void* d_ws, size_t ws_size,
                              hipStream_t stream) {
    const float* x     = (const float*)d_in[0];   // 1024x128
    const float* adj   = (const float*)d_in[1];   // 1024x1024
    const float* w     = (const float*)d_in[2];   // 128x128
    const float* bias  = (const float*)d_in[3];   // 128
    const float* att_w = (const float*)d_in[4];   // 256
    const float* att_b = (const float*)d_in[5];   // 1

    float* ws   = (float*)d_ws;
    float* mv   = ws;                 // 1024*128
    float* h    = ws + 131072;        // 1024*128
    float* supp = ws + 262144;        // 1024*128
    float* x2a  = ws + 393216;        // 1024
    float* tla  = ws + 394240;        // 1024
    float* tra  = ws + 395264;        // 1024
    float* out  = (float*)d_out;

    k_linear_wmma<<<dim3(64, 8), 32, 0, stream>>>(x, w, mv);
    k_rowprep<<<1024, 128, 0, stream>>>(x, mv, bias, att_w, h, x2a, tla, tra);
    k_att_agg<<<64, 32, 0, stream>>>(h, adj, x2a, tla, tra, att_b, supp);
    k_finish<<<1024, 128, 0, stream>>>(h, supp, x2a, out);
}